// EnhancedPhysicsLoss_32066225832268
// MI455X (gfx1250) — compile-verified
//
#include <hip/hip_runtime.h>
#include <hip/hip_bf16.h>
#include <stdint.h>

// ---------------------------------------------------------------------------
// EnhancedPhysicsLoss for MI455X (gfx1250, wave32)
//
// zero_ws -> edge_kernel (async global->LDS double-buffered streaming of the
// edge tables, L2-resident global_atomic_add_f32 segment sums) ->
// node_kernel (per-node losses, v_wmma_f32_16x16x4_f32 block reduction) ->
// final_kernel (weighted combine).
//
// step input is ignored: CONT_INIT==CONT_TGT and MOM_INIT==MOM_TGT, so the
// curriculum weights are 0.1 for every step value.
// ---------------------------------------------------------------------------

typedef float v2f __attribute__((ext_vector_type(2)));
typedef float v8f __attribute__((ext_vector_type(8)));

#define NNODES 100000
#define NEDGES 6400000
#define TILE 512                    // edges per block tile (2 per lane)
#define NUM_TILES (NEDGES / TILE)   // 12500 exact, no tail
#define EPSF 1e-8f
#define NU_MOL 1.5e-5f

// ---------------------------------------------------------------------------
__global__ __launch_bounds__(256) void zero_ws_kernel(float* __restrict__ ws, int n) {
  int i = blockIdx.x * blockDim.x + threadIdx.x;
  if (i < n) ws[i] = 0.0f;
}

// ---------------------------------------------------------------------------
// Edge pass. Each lane owns edges (tileBase + 2*tid, +1). Per tile each wave
// issues 3 async loads (rows b64, cols b64, dx/dy b128); ASYNCcnt is
// per-wave with in-order completion, and every wave reads only the LDS slots
// it wrote, so s_wait_asynccnt alone orders producer/consumer (no barrier).
// Double buffer: prefetch tile i+1 while computing tile i.
// ---------------------------------------------------------------------------
__global__ __launch_bounds__(256) void edge_kernel(
    const float4* __restrict__ pred,
    const float2* __restrict__ attr,
    const int* __restrict__ erow,
    const int* __restrict__ ecol,
    float* __restrict__ w_cnt, float* __restrict__ w_div,
    float* __restrict__ w_pgx, float* __restrict__ w_pgy,
    float* __restrict__ w_lapu, float* __restrict__ w_lapv,
    float* __restrict__ w_str, float* __restrict__ scal) {
  __shared__ __align__(16) int s_row[2][TILE];
  __shared__ __align__(16) int s_col[2][TILE];
  __shared__ __align__(16) float2 s_att[2][TILE];
  __shared__ float s_sm[8];

  const int tid = threadIdx.x;
  const int stride = (int)gridDim.x;
  float smooth_acc = 0.0f;

  auto issue = [&](int t, int b) {
    long long e = (long long)t * TILE + 2 * tid;           // 2 edges per lane
    unsigned lr = (unsigned)(uintptr_t)&s_row[b][2 * tid]; // low 32 bits of flat
    unsigned lc = (unsigned)(uintptr_t)&s_col[b][2 * tid]; // LDS pointer == LDS
    unsigned la = (unsigned)(uintptr_t)&s_att[b][2 * tid]; // byte offset
    unsigned long long gr = (unsigned long long)(uintptr_t)(erow + e);
    unsigned long long gc = (unsigned long long)(uintptr_t)(ecol + e);
    unsigned long long ga = (unsigned long long)(uintptr_t)(attr + e);
    asm volatile("global_load_async_to_lds_b64 %0, %1, off" ::"v"(lr), "v"(gr) : "memory");
    asm volatile("global_load_async_to_lds_b64 %0, %1, off" ::"v"(lc), "v"(gc) : "memory");
    asm volatile("global_load_async_to_lds_b128 %0, %1, off" ::"v"(la), "v"(ga) : "memory");
  };

  int t = blockIdx.x;
  int buf = 0;
  if (t < NUM_TILES) issue(t, 0);
  for (; t < NUM_TILES; t += stride, buf ^= 1) {
    int tn = t + stride;
    if (tn < NUM_TILES) {
      issue(tn, buf ^ 1);                                // prefetch next tile
      asm volatile("s_wait_asynccnt 0x3" ::: "memory");  // current tile's 3 done
    } else {
      asm volatile("s_wait_asynccnt 0x0" ::: "memory");
    }
#pragma unroll
    for (int s = 0; s < 2; ++s) {
      int slot = 2 * tid + s;
      int r = s_row[buf][slot];
      int c = s_col[buf][slot];
      float2 a = s_att[buf][slot];
      float4 pr = pred[r];
      float4 pc = pred[c];
      float du = pc.x - pr.x;
      float dv = pc.y - pr.y;
      float dp = pc.z - pr.z;
      float dn = pc.w - pr.w;
      float idx = 1.0f / (a.x + EPSF);
      float idy = 1.0f / (a.y + EPSF);
      float idx2 = 1.0f / (a.x * a.x + EPSF);
      float idy2 = 1.0f / (a.y * a.y + EPSF);
      float dudx = du * idx, dudy = du * idy;
      float dvdx = dv * idx, dvdy = dv * idy;
      unsafeAtomicAdd(&w_cnt[r], 1.0f);
      unsafeAtomicAdd(&w_div[r], dudx + dvdy);
      unsafeAtomicAdd(&w_pgx[r], dp * idx);
      unsafeAtomicAdd(&w_pgy[r], dp * idy);
      unsafeAtomicAdd(&w_lapu[r], du * idx2);
      unsafeAtomicAdd(&w_lapv[r], dv * idy2);
      float sdt = dudy + dvdx;
      unsafeAtomicAdd(&w_str[r], 2.0f * (dudx * dudx + dvdy * dvdy) + sdt * sdt);
      smooth_acc += du * du + dv * dv + dp * dp + dn * dn;
    }
  }

  // Block-reduce the smoothness partial sum, single atomic per block.
  float v = smooth_acc;
#pragma unroll
  for (int off = 16; off >= 1; off >>= 1) v += __shfl_xor(v, off, 32);
  int wave = tid >> 5, lane = tid & 31;
  if (lane == 0) s_sm[wave] = v;
  __syncthreads();
  if (tid == 0) {
    float s = 0.0f;
#pragma unroll
    for (int w2 = 0; w2 < 8; ++w2) s += s_sm[w2];
    unsafeAtomicAdd(&scal[8], s);
  }
}

// ---------------------------------------------------------------------------
// Node pass: 8 scalar loss quantities per node, reduced per-wave with
// shuffles, then across the 8 waves of the block with v_wmma_f32_16x16x4_f32:
// A[q][w] = wave w's partial of quantity q (A 16x4 layout: lanes 0-15 carry
// K=0,1; lanes 16-31 carry K=2,3), B = all-ones (layout-insensitive), so
// D[q][*] = sum over waves. Two chained WMMAs cover all 8 waves. The LDS
// staging tile is padded to 16 rows of zeros so the A loads are unconditional
// straight-line ds_loads (no exec-mask branch regions).
// ---------------------------------------------------------------------------
__global__ __launch_bounds__(256) void node_kernel(
    const float4* __restrict__ pred, const float4* __restrict__ tgt,
    const float* __restrict__ w_cnt, const float* __restrict__ w_div,
    const float* __restrict__ w_pgx, const float* __restrict__ w_pgy,
    const float* __restrict__ w_lapu, const float* __restrict__ w_lapv,
    const float* __restrict__ w_str, const unsigned char* __restrict__ wall,
    float* __restrict__ scal) {
  int i = blockIdx.x * blockDim.x + threadIdx.x;
  float q[8] = {0.f, 0.f, 0.f, 0.f, 0.f, 0.f, 0.f, 0.f};
  if (i < NNODES) {
    float4 p = pred[i];
    float4 g = tgt[i];
    float inv = 1.0f / fmaxf(w_cnt[i], 1.0f);
    float d0 = p.x - g.x, d1 = p.y - g.y, d2 = p.z - g.z, d3 = p.w - g.w;
    q[0] = d0 * d0 + d1 * d1 + d2 * d2 + d3 * d3;           // mse sum
    float dvg = w_div[i] * inv;
    q[1] = dvg * dvg;                                       // divergence^2
    float nut = p.w;
    float nueff = NU_MOL + nut;
    float mx = w_pgx[i] * inv + nueff * (w_lapu[i] * inv);
    float my = w_pgy[i] * inv + nueff * (w_lapv[i] * inv);
    q[2] = mx * mx + my * my;                               // momentum
    float prodv = nut * (w_str[i] * inv);
    q[3] = prodv * prodv;                                   // production^2
    q[4] = nut * nut;                                       // dissipation
    float m = wall[i] ? 1.0f : 0.0f;
    float uv = p.x * p.x + p.y * p.y;
    q[5] = m * uv;                                          // bc sum
    q[6] = m * (uv + nut * nut);                            // wall sum
    q[7] = m;                                               // n_wall
  }
  // wave32 xor-reduction (all lanes end with the wave total)
#pragma unroll
  for (int k = 0; k < 8; ++k) {
    float v = q[k];
#pragma unroll
    for (int off = 16; off >= 1; off >>= 1) v += __shfl_xor(v, off, 32);
    q[k] = v;
  }
  __shared__ float s_part[8][16]; // [wave][quantity], rows 8..15 zero-padded
  int wave = threadIdx.x >> 5, lane = threadIdx.x & 31;
  if (lane == 0) {
#pragma unroll
    for (int k = 0; k < 8; ++k) s_part[wave][k] = q[k];
#pragma unroll
    for (int k = 8; k < 16; ++k) s_part[wave][k] = 0.0f;
  }
  __syncthreads();
  if (wave == 0) { // wave-uniform branch: EXEC all-ones for WMMA
    int row = lane & 15;
    int k0 = (lane < 16) ? 0 : 2; // A layout: lanes<16 -> K=0,1 ; lanes>=16 -> K=2,3
    v2f a0, a1, b;
    b.x = 1.0f;
    b.y = 1.0f;
    a0.x = s_part[k0 + 0][row];
    a0.y = s_part[k0 + 1][row];
    a1.x = s_part[k0 + 4][row];
    a1.y = s_part[k0 + 5][row];
    v8f acc = {};
    acc = __builtin_amdgcn_wmma_f32_16x16x4_f32(false, a0, false, b, (short)0, acc, false, false);
    acc = __builtin_amdgcn_wmma_f32_16x16x4_f32(false, a1, false, b, (short)0, acc, false, false);
    // D layout: VGPR j, lanes 0-15 hold M=j (replicated over N), so every
    // lane < 16 holds all 8 block sums in acc[0..7]. Lanes 0-7 each commit
    // one quantity; non-uniform address keeps this a single vector atomic.
    if (lane < 8) {
      float v = acc[0];
#pragma unroll
      for (int k = 1; k < 8; ++k) v = (lane == k) ? acc[k] : v;
      unsafeAtomicAdd(&scal[lane], v);
    }
  }
}

// ---------------------------------------------------------------------------
__global__ void final_kernel(const float* __restrict__ scal, float* __restrict__ out) {
  if (threadIdx.x == 0 && blockIdx.x == 0) {
    const float invN = 1.0f / (float)NNODES;
    float mse = scal[0] * (1.0f / (4.0f * (float)NNODES));
    float divq = scal[1] * invN;
    float mom = scal[2] * invN;
    float prod = scal[3] * invN;
    float diss = scal[4] * invN;
    float nwall = fmaxf(scal[7], 1.0f);
    float bc = scal[5] / nwall;
    float wallv = scal[6] / nwall;
    float smooth = scal[8] * (1.0f / (4.0f * (float)NEDGES));
    // DATA_W=1, cont_w=mom_w=0.1 (constant for all steps), TURB_W=RANS_W=0.05,
    // BC_W=0.05, SMOOTH_W=0.01, WALL_W=0.02
    out[0] = 1.0f * mse + 0.1f * divq + 0.1f * mom + 0.05f * prod +
             0.05f * diss + 0.05f * bc + 0.01f * smooth + 0.02f * wallv;
  }
}

// ---------------------------------------------------------------------------
extern "C" void kernel_launch(void* const* d_in, const int* in_sizes, int n_in,
                              void* d_out, int out_size, void* d_ws, size_t ws_size,
                              hipStream_t stream) {
  const float4* pred = (const float4*)d_in[0];               // (N,4) f32
  const float4* tgt = (const float4*)d_in[1];                // (N,4) f32
  const float2* attr = (const float2*)d_in[2];               // (E,2) f32
  const int* eidx = (const int*)d_in[3];                     // (2,E) int32
  const unsigned char* wall = (const unsigned char*)d_in[4]; // (N,) bool
  // d_in[5] = step: unused, curriculum weights are constant (INIT==TGT).

  float* ws = (float*)d_ws; // needs 7*N + 16 floats = ~2.8 MB
  float* w_cnt = ws + 0 * NNODES;
  float* w_div = ws + 1 * NNODES;
  float* w_pgx = ws + 2 * NNODES;
  float* w_pgy = ws + 3 * NNODES;
  float* w_lapu = ws + 4 * NNODES;
  float* w_lapv = ws + 5 * NNODES;
  float* w_str = ws + 6 * NNODES;
  float* scal = ws + 7 * NNODES; // [0..8] scalar accumulators

  int nz = 7 * NNODES + 16;
  zero_ws_kernel<<<(nz + 255) / 256, 256, 0, stream>>>(ws, nz);

  edge_kernel<<<2048, 256, 0, stream>>>(pred, attr, eidx, eidx + NEDGES,
                                        w_cnt, w_div, w_pgx, w_pgy,
                                        w_lapu, w_lapv, w_str, scal);

  node_kernel<<<(NNODES + 255) / 256, 256, 0, stream>>>(pred, tgt, w_cnt, w_div,
                                                        w_pgx, w_pgy, w_lapu,
                                                        w_lapv, w_str, wall, scal);

  final_kernel<<<1, 32, 0, stream>>>(scal, (float*)d_out);
}